// DistanceMatrix_8598524527263
// MI455X (gfx1250) — compile-verified
//
#include <hip/hip_runtime.h>
#include <stdint.h>

// Problem constants from the reference (B,H,W fixed by setup_inputs)
#define B_ 8
#define H_ 256
#define W_ 256
#define LARGE_F 131072.0f   // H*H + W*W

typedef unsigned int v4u __attribute__((ext_vector_type(4)));
typedef int          v4i __attribute__((ext_vector_type(4)));
typedef int          v8i __attribute__((ext_vector_type(8)));

// ---------------------------------------------------------------------------
// Tensor Data Mover: load a 2-D fp32 tile (tile_w x tile_h elements, row
// stride stride_elems) from global memory into LDS, per CDNA5 D# descriptor
// layout (CDNA5 ISA).
// ---------------------------------------------------------------------------
__device__ __forceinline__ void tdm_load_2d_f32(const void* gptr, void* ldsptr,
                                                unsigned tile_w, unsigned tile_h,
                                                unsigned tensor_w, unsigned tensor_h,
                                                unsigned stride_elems) {
  unsigned long long ga = (unsigned long long)(uintptr_t)gptr;
  unsigned lds_off = (unsigned)(uintptr_t)ldsptr;  // low 32 bits of generic LDS ptr = offset

  // D# group 0: [1:0]=count=1, [63:32]=lds_addr, [120:64]=global_addr, [127:126]=type=2
  v4u g0;
  g0.x = 1u;
  g0.y = lds_off;
  g0.z = (unsigned)ga;
  g0.w = ((unsigned)(ga >> 32) & 0x01FFFFFFu) | (2u << 30);

  // D# group 1: data_size[17:16]=2 (4B); tensor_dim0 @[79:48]; tensor_dim1 @[111:80];
  // tile_dim0 @[127:112]; tile_dim1 @[143:128]; tensor_dim0_stride @[207:160]
  v8i g1;
  g1[0] = (int)(2u << 16);
  g1[1] = (int)((tensor_w & 0xFFFFu) << 16);
  g1[2] = (int)((tensor_w >> 16) | ((tensor_h & 0xFFFFu) << 16));
  g1[3] = (int)((tensor_h >> 16) | (tile_w << 16));
  g1[4] = (int)tile_h;          // tile_dim1; tile_dim2 = 0 (unused -> 2D)
  g1[5] = (int)stride_elems;    // tensor_dim0_stride[31:0]
  g1[6] = 0;
  g1[7] = 0;

  v4i zero4 = {0, 0, 0, 0};
#if __clang_major__ >= 23
  v8i zero8 = {0, 0, 0, 0, 0, 0, 0, 0};
  __builtin_amdgcn_tensor_load_to_lds(g0, g1, zero4, zero4, zero8, 0);
#else
  __builtin_amdgcn_tensor_load_to_lds(g0, g1, zero4, zero4, 0);
#endif
}

// ---------------------------------------------------------------------------
// Pass 1: per-column 1-D distance transform of the binary mask.
// One wave (32 threads) per block; each block handles 32 columns of one image.
// TDM stages a 256x32 tile into LDS; forward scan overwrites mask with the
// forward distance in-place; backward scan combines and writes d1 = min(n^2, L).
// ---------------------------------------------------------------------------
__global__ __launch_bounds__(32) void dt_pass1(const float* __restrict__ mask,
                                               float* __restrict__ out) {
  __shared__ float tile[H_ * 32];
  const int b  = blockIdx.x >> 3;         // 8 column-tiles per image
  const int y0 = (blockIdx.x & 7) << 5;   // first column of this tile
  const int t  = threadIdx.x;             // column within tile

  // Stage mask[b, 0:256, y0:y0+32] -> tile[x*32 + c] via Tensor Data Mover.
  tdm_load_2d_f32(mask + (size_t)b * H_ * W_ + y0, tile,
                  /*tile_w=*/32, /*tile_h=*/H_,
                  /*tensor_w=*/32, /*tensor_h=*/H_, /*stride=*/W_);
  __builtin_amdgcn_s_wait_tensorcnt(0);   // single-wave block: no barrier needed

  // Forward scan: distance to nearest foreground above (0 on foreground).
  float fwd = 10000.0f;
#pragma unroll 4
  for (int x = 0; x < H_; ++x) {
    float m = tile[x * 32 + t];
    fwd = (m > 0.5f) ? 0.0f : (fwd + 1.0f);
    tile[x * 32 + t] = fwd;               // overwrite mask with fwd distance
  }

  // Backward scan: combine, square, clamp, write d1 (coalesced across lanes).
  float bwd = 10000.0f;
  float* col = out + (size_t)b * H_ * W_ + y0 + t;
#pragma unroll 4
  for (int x = H_ - 1; x >= 0; --x) {
    float fv = tile[x * 32 + t];
    bwd = (fv == 0.0f) ? 0.0f : (bwd + 1.0f);   // fv==0 <=> foreground
    float d = fminf(fv, bwd);
    col[(size_t)x * W_] = fminf(d * d, LARGE_F);
  }
}

// ---------------------------------------------------------------------------
// Pass 2: per-row min-plus with the parabola kernel, brute force O(W) per
// output. 256 threads / block, 8 rows / block, in-place on d_out.
// TDM stages the 8x256 row tile; LDS reads are wave-uniform broadcasts.
// ---------------------------------------------------------------------------
__global__ __launch_bounds__(256) void dt_pass2(float* __restrict__ out) {
  __shared__ float rows[8 * W_];
  const int row0 = blockIdx.x * 8;        // global row index in [0, B*H)
  const int j    = threadIdx.x;           // output column

  if (j < 32) {                            // wave 0 issues the tensor DMA
    tdm_load_2d_f32(out + (size_t)row0 * W_, rows,
                    /*tile_w=*/W_, /*tile_h=*/8,
                    /*tensor_w=*/W_, /*tensor_h=*/8, /*stride=*/W_);
    __builtin_amdgcn_s_wait_tensorcnt(0);
  }
  __syncthreads();

  const float fj = (float)j;
  for (int r = 0; r < 8; ++r) {
    const float* row = &rows[r * W_];
    float m = LARGE_F;                     // final cap min(d2, LARGE) folded in
#pragma unroll 8
    for (int y = 0; y < W_; ++y) {
      float dy = fj - (float)y;
      m = fminf(m, fmaf(dy, dy, row[y]));  // dy*dy exact (small ints)
    }
    out[(size_t)(row0 + r) * W_ + j] = sqrtf(m) * 0.1f;
  }
}

// ---------------------------------------------------------------------------
extern "C" void kernel_launch(void* const* d_in, const int* in_sizes, int n_in,
                              void* d_out, int out_size, void* d_ws, size_t ws_size,
                              hipStream_t stream) {
  (void)in_sizes; (void)n_in; (void)out_size; (void)d_ws; (void)ws_size;
  const float* mask = (const float*)d_in[0];
  float* out = (float*)d_out;

  dt_pass1<<<dim3(B_ * (W_ / 32)), dim3(32), 0, stream>>>(mask, out);   // 64 blocks
  dt_pass2<<<dim3(B_ * H_ / 8),    dim3(256), 0, stream>>>(out);        // 256 blocks
}